// GNN_70145405878616
// MI455X (gfx1250) — compile-verified
//
#include <hip/hip_runtime.h>

#ifndef __has_builtin
#define __has_builtin(x) 0
#endif

// gfx1250 v_tanh_f32 when available.
__device__ __forceinline__ float gtanh(float x) {
#if __has_builtin(__builtin_amdgcn_tanhf)
  return __builtin_amdgcn_tanhf(x);
#else
  return tanhf(x);
#endif
}

typedef float v4f __attribute__((ext_vector_type(4)));

#define WAVES_PER_BLOCK 8
#define THREADS (WAVES_PER_BLOCK * 32)
#define PAIR_BYTES 1536u  // 2 nodes * 3 ch * 64 floats
#define CHUNKS 3          // 3 x (32 lanes x 16 B) = 1536 B per pair

// Low 32 bits of the flat (generic) address of an LDS object are the DS byte
// offset (LDS_BASE is added by hardware).
__device__ __forceinline__ uint32_t lds_off(const void* p) {
  return (uint32_t)(uintptr_t)p;
}

// Streaming GNN kernel on the CDNA5 async global->LDS engine.
// Each wave double-buffers one node pair (1536 B) per iteration:
//   issue GVS async loads for pair n+1, s_wait_asynccnt 0x3 (in-order
//   completion => pair n's 3 loads done), reduce pair n from LDS.
// No barriers: every wave consumes only data it loaded itself.
__global__ __launch_bounds__(THREADS) void gnn_async_kernel(
    const float* __restrict__ x,    // [B,3]
    const float* __restrict__ nbr,  // [B,3,64]
    const float* __restrict__ Ws,   // [3,3]
    const float* __restrict__ bs,   // [3]
    const float* __restrict__ Wc,   // [3,3,2]
    const float* __restrict__ bc,   // [3]
    float* __restrict__ out,        // [B,3]
    int B, int totalPairs) {
  __shared__ __align__(16) float smem[WAVES_PER_BLOCK][2][PAIR_BYTES / 4];

  const int tid  = threadIdx.x;
  const int w    = tid >> 5;
  const int lane = tid & 31;

  const uint32_t waveGlobal = blockIdx.x * WAVES_PER_BLOCK + w;
  const uint32_t waveStride = gridDim.x * WAVES_PER_BLOCK;
  const uint32_t nBytes     = (uint32_t)B * 768u;  // B*192 floats; fits 32-bit (768 MB)

  const uint64_t nbrBase  = (uint64_t)(uintptr_t)nbr;                 // SGPR pair (GVS saddr)
  const uint32_t laneByte = (uint32_t)lane * 16u;
  const uint32_t ldsBase  = lds_off(&smem[w][0][0]) + laneByte;

  // Issue 3 async B128 loads (512 B each, coalesced) for one pair into buffer `buf`.
  auto issue = [&](uint32_t pair, int buf) {
    const uint32_t pairByte = pair * PAIR_BYTES + laneByte;
#pragma unroll
    for (int t = 0; t < CHUNKS; ++t) {
      uint32_t goff = pairByte + (uint32_t)t * 512u;
      if (goff + 16u > nBytes) goff = 0;  // clamp tail lanes to a safe address
      const uint32_t laddr = ldsBase + (uint32_t)buf * PAIR_BYTES + (uint32_t)t * 512u;
      asm volatile("global_load_async_to_lds_b128 %0, %1, %2 th:TH_LOAD_NT"
                   :: "v"(laddr), "v"(goff), "s"(nbrBase)
                   : "memory");
    }
  };

  // Reduce one node pair out of LDS buffer `buf` (half-wave per node).
  auto compute = [&](uint32_t pair, int buf) {
    const int half = lane >> 4;
    const int l16  = lane & 15;
    const uint32_t node = pair * 2u + (uint32_t)half;

    const float* sb = &smem[w][buf][half * 192];
    v4f a = *(const v4f*)(sb + l16 * 4);        // channel 0
    v4f b = *(const v4f*)(sb + 64 + l16 * 4);   // channel 1
    v4f c = *(const v4f*)(sb + 128 + l16 * 4);  // channel 2

    float s0 = (a.x + a.y) + (a.z + a.w);
    float s1 = (b.x + b.y) + (b.z + b.w);
    float s2 = (c.x + c.y) + (c.z + c.w);

#pragma unroll
    for (int off = 8; off >= 1; off >>= 1) {
      s0 += __shfl_xor(s0, off, 32);
      s1 += __shfl_xor(s1, off, 32);
      s2 += __shfl_xor(s2, off, 32);
    }

    if (node < (uint32_t)B && l16 < 3) {
      const float inv = 1.0f / 64.0f;
      const float p0 = s0 * inv, p1 = s1 * inv, p2 = s2 * inv;

      const float x0 = __builtin_nontemporal_load(x + node * 3 + 0);
      const float x1 = __builtin_nontemporal_load(x + node * 3 + 1);
      const float x2 = __builtin_nontemporal_load(x + node * 3 + 2);

      const float h0 = gtanh(fmaf(Ws[0], x0, fmaf(Ws[1], x1, fmaf(Ws[2], x2, bs[0]))));
      const float h1 = gtanh(fmaf(Ws[3], x0, fmaf(Ws[4], x1, fmaf(Ws[5], x2, bs[1]))));
      const float h2 = gtanh(fmaf(Ws[6], x0, fmaf(Ws[7], x1, fmaf(Ws[8], x2, bs[2]))));

      const int o = l16;
      float z = bc[o];
      z = fmaf(Wc[o * 6 + 0], h0, z);
      z = fmaf(Wc[o * 6 + 1], p0, z);
      z = fmaf(Wc[o * 6 + 2], h1, z);
      z = fmaf(Wc[o * 6 + 3], p1, z);
      z = fmaf(Wc[o * 6 + 4], h2, z);
      z = fmaf(Wc[o * 6 + 5], p2, z);

      __builtin_nontemporal_store(gtanh(gtanh(z)), out + node * 3 + o);
    }
  };

  uint32_t p = waveGlobal;
  if (p >= (uint32_t)totalPairs) return;  // wave-uniform

  issue(p, 0);
  int buf = 0;
  for (;;) {
    const uint32_t pn = p + waveStride;
    const bool more = pn < (uint32_t)totalPairs;  // wave-uniform
    if (more) {
      issue(pn, buf ^ 1);
      // 6 outstanding; async loads complete in order => cnt<=3 means pair p is in LDS.
      asm volatile("s_wait_asynccnt 0x3" ::: "memory");
    } else {
      asm volatile("s_wait_asynccnt 0x0" ::: "memory");
    }
    compute(p, buf);
    if (!more) break;
    p = pn;
    buf ^= 1;
  }
}

extern "C" void kernel_launch(void* const* d_in, const int* in_sizes, int n_in,
                              void* d_out, int out_size, void* d_ws, size_t ws_size,
                              hipStream_t stream) {
  const float* x   = (const float*)d_in[0];
  const float* nbr = (const float*)d_in[1];
  const float* Ws  = (const float*)d_in[2];
  const float* bs  = (const float*)d_in[3];
  const float* Wc  = (const float*)d_in[4];
  const float* bc  = (const float*)d_in[5];
  float* out = (float*)d_out;

  const int B = in_sizes[0] / 3;  // x is [B,3,1]
  const int totalPairs = (B + 1) / 2;

  int blocks = (totalPairs + WAVES_PER_BLOCK - 1) / WAVES_PER_BLOCK;
  if (blocks > 4096) blocks = 4096;  // grid-stride: ~15 pairs/wave => real double-buffering

  gnn_async_kernel<<<blocks, THREADS, 0, stream>>>(x, nbr, Ws, bs, Wc, bc, out, B, totalPairs);
}